// DGI_24455543783861
// MI455X (gfx1250) — compile-verified
//
#include <hip/hip_runtime.h>
#include <hip/hip_bf16.h>

typedef __attribute__((ext_vector_type(2))) float v2f;
typedef __attribute__((ext_vector_type(8))) float v8f;

#define HIDW 32   // hidden width is fixed at 32 in this problem (pooling/GEMM lane maps assume it)

// ---------------- degree / normalization ----------------
__global__ void k_fill_deg(float* deg, int n) {
    int i = blockIdx.x * blockDim.x + threadIdx.x;
    if (i < n) deg[i] = 1.0f;                       // self-loop contributes 1
}

__global__ void k_count_deg(const int* __restrict__ dst, float* deg, int e) {
    int i = blockIdx.x * blockDim.x + threadIdx.x;
    if (i < e) atomicAdd(&deg[dst[i]], 1.0f);
}

__global__ void k_dinv(float* deg, int n) {
    int i = blockIdx.x * blockDim.x + threadIdx.x;
    if (i < n) {
        float d = deg[i];
        deg[i] = (d > 0.0f) ? rsqrtf(d) : 0.0f;     // in-place deg -> d^{-1/2}
    }
}

// ---------------- WMMA f32 GEMM: Y[nrows,32] = X[nrows,K] @ W[K,32] ----------------
// One wave per 16-row tile; two 16x16 accumulators cover the 32 output columns.
// A layout (16x4 f32): lanes 0-15 row M=lane hold K={k,k+1}; lanes 16-31 hold K={k+2,k+3}.
// B layout (4x16 f32): VGPR0 = rows k (lanes 0-15) / k+2 (lanes 16-31); VGPR1 = k+1 / k+3.
__global__ void k_gemm_wmma(const float* __restrict__ X, const float* __restrict__ W,
                            float* __restrict__ Y, int nrows, int K) {
    int gid  = blockIdx.x * blockDim.x + threadIdx.x;
    int wave = gid >> 5;                 // blockDim multiple of 32 -> wave-uniform
    int lane = threadIdx.x & 31;
    int row0 = wave << 4;
    if (row0 >= nrows) return;           // wave-uniform branch: EXEC stays all-ones
    int half = lane >> 4;                // 0: K pair {0,1}, 1: K pair {2,3}
    int l16  = lane & 15;

    v8f acc0 = {};                       // columns 0..15
    v8f acc1 = {};                       // columns 16..31
    const float* xrow  = X + (size_t)(row0 + l16) * K + 2 * half;
    const float* wbase = W + (size_t)(2 * half) * HIDW + l16;

    for (int k = 0; k < K; k += 4) {
        v2f a = *(const v2f*)(xrow + k);                 // b64, contiguous pair
        const float* wp = wbase + (size_t)k * HIDW;
        v2f b0 = { wp[0],  wp[HIDW] };                   // rows k(+2h), k+1(+2h), col l16
        v2f b1 = { wp[16], wp[HIDW + 16] };              // same, cols 16..31
        acc0 = __builtin_amdgcn_wmma_f32_16x16x4_f32(false, a, false, b0,
                                                     (short)0, acc0, false, false);
        acc1 = __builtin_amdgcn_wmma_f32_16x16x4_f32(false, a, false, b1,
                                                     (short)0, acc1, false, false);
    }

    // C/D layout: VGPR j -> row (j + 8*half), col l16 (+16 for acc1)
    float* yb = Y + (size_t)(row0 + 8 * half) * HIDW + l16;
#pragma unroll
    for (int j = 0; j < 8; ++j) {
        yb[(size_t)j * HIDW]      = acc0[j];
        yb[(size_t)j * HIDW + 16] = acc1[j];
    }
}

// ---------------- GCN propagation ----------------
// out[n,:] = h[n,:] * dinv[n]^2   (self-loop term doubles as zero-init)
__global__ void k_init_self(const float* __restrict__ h, const float* __restrict__ dinv,
                            float* __restrict__ out, int nelem) {
    int i = blockIdx.x * blockDim.x + threadIdx.x;
    if (i >= nelem) return;
    float dv = dinv[i >> 5];             // HIDW == 32
    out[i] = h[i] * dv * dv;
}

// one wave per edge; 32 lanes = 32 channels; coalesced 128B gather + 128B atomic
__global__ void k_edge_prop(const int* __restrict__ src, const int* __restrict__ dst,
                            const float* __restrict__ h, const float* __restrict__ dinv,
                            float* __restrict__ out, int nedges) {
    int gid  = blockIdx.x * blockDim.x + threadIdx.x;
    int e    = gid >> 5;
    int lane = threadIdx.x & 31;
    if (e >= nedges) return;
    int s = src[e];
    int d = dst[e];
    float norm = dinv[s] * dinv[d];
    float v = h[(size_t)s * HIDW + lane] * norm;
    atomicAdd(&out[(size_t)d * HIDW + lane], v);
}

__global__ void k_bias_act(float* out, const float* __restrict__ bias, int nelem, int do_relu) {
    int i = blockIdx.x * blockDim.x + threadIdx.x;
    if (i >= nelem) return;
    float v = out[i] + bias[i & (HIDW - 1)];
    if (do_relu) v = fmaxf(v, 0.0f);
    out[i] = v;
}

// ---------------- pooling ----------------
__global__ void k_zero(float* p, int n) {
    int i = blockIdx.x * blockDim.x + threadIdx.x;
    if (i < n) p[i] = 0.0f;
}

__global__ void k_pool(const float* __restrict__ h, const int* __restrict__ batch,
                       float* gsum, float* cnt, int nnodes) {
    int gid  = blockIdx.x * blockDim.x + threadIdx.x;
    int node = gid >> 5;
    int lane = threadIdx.x & 31;
    if (node >= nnodes) return;
    int g = batch[node];
    atomicAdd(&gsum[(size_t)g * HIDW + lane], h[(size_t)node * HIDW + lane]);
    if (lane == 0) atomicAdd(&cnt[g], 1.0f);
}

__global__ void k_gdiv(float* gsum, const float* __restrict__ cnt, int nelem) {
    int i = blockIdx.x * blockDim.x + threadIdx.x;
    if (i >= nelem) return;
    gsum[i] = gsum[i] / fmaxf(cnt[i >> 5], 1.0f);
}

static inline int cdiv_i(int a, int b) { return (a + b - 1) / b; }

extern "C" void kernel_launch(void* const* d_in, const int* in_sizes, int n_in,
                              void* d_out, int out_size, void* d_ws, size_t ws_size,
                              hipStream_t stream) {
    const float* x   = (const float*)d_in[0];
    const int*   ei  = (const int*)d_in[1];
    const int*   bat = (const int*)d_in[2];
    const float* W1  = (const float*)d_in[3];
    const float* b1  = (const float*)d_in[4];
    const float* W2  = (const float*)d_in[5];
    const float* b2  = (const float*)d_in[6];
    const float* Wsm = (const float*)d_in[7];
    const float* bsm = (const float*)d_in[8];
    float* out = (float*)d_out;

    const int HID = HIDW;
    const int INC = in_sizes[3] / HID;                 // 128
    const int N   = in_sizes[0] / INC;                 // 100000
    const int E   = in_sizes[1] / 2;                   // 1600000
    const int G   = (out_size - N * HID) / HID;        // 64
    const int* src = ei;
    const int* dst = ei + E;

    // workspace layout (floats)
    float* ws   = (float*)d_ws;
    float* bufA = ws;                                  // N*HID  (GEMM outputs h)
    float* bufB = bufA + (size_t)N * HID;              // N*HID  (layer-1 propagated)
    float* dinv = bufB + (size_t)N * HID;              // N      (deg -> d^{-1/2} in place)
    float* gsum = dinv + N;                            // G*HID
    float* cnt  = gsum + (size_t)G * HID;              // G   (contiguous with gsum for zeroing)

    float* summary = out;                              // [G,HID]
    float* hout    = out + (size_t)G * HID;            // [N,HID] node embeddings

    const int B = 256;
    const int tilesN = cdiv_i(N, 16);
    const int tilesG = cdiv_i(G, 16);

    // degrees + normalization (shared by both conv layers)
    k_fill_deg <<<cdiv_i(N, B), B, 0, stream>>>(dinv, N);
    k_count_deg<<<cdiv_i(E, B), B, 0, stream>>>(dst, dinv, E);
    k_dinv     <<<cdiv_i(N, B), B, 0, stream>>>(dinv, N);

    // ---- layer 1: h1 = x@W1 ; propagate ; +b1 ; relu ----
    k_gemm_wmma<<<cdiv_i(tilesN * 32, B), B, 0, stream>>>(x, W1, bufA, N, INC);
    k_init_self<<<cdiv_i(N * HID, B), B, 0, stream>>>(bufA, dinv, bufB, N * HID);
    k_edge_prop<<<cdiv_i(E * 32, B), B, 0, stream>>>(src, dst, bufA, dinv, bufB, E);
    k_bias_act <<<cdiv_i(N * HID, B), B, 0, stream>>>(bufB, b1, N * HID, 1);

    // ---- layer 2: h2 = relu1@W2 ; propagate into d_out ; +b2 ----
    k_gemm_wmma<<<cdiv_i(tilesN * 32, B), B, 0, stream>>>(bufB, W2, bufA, N, HID);
    k_init_self<<<cdiv_i(N * HID, B), B, 0, stream>>>(bufA, dinv, hout, N * HID);
    k_edge_prop<<<cdiv_i(E * 32, B), B, 0, stream>>>(src, dst, bufA, dinv, hout, E);
    k_bias_act <<<cdiv_i(N * HID, B), B, 0, stream>>>(hout, b2, N * HID, 0);

    // ---- global mean pool ----
    k_zero<<<cdiv_i(G * HID + G, B), B, 0, stream>>>(gsum, G * HID + G);
    k_pool<<<cdiv_i(N * 32, B), B, 0, stream>>>(hout, bat, gsum, cnt, N);
    k_gdiv<<<cdiv_i(G * HID, B), B, 0, stream>>>(gsum, cnt, G * HID);

    // ---- summary = g @ Ws + bs ----
    k_gemm_wmma<<<cdiv_i(tilesG * 32, B), B, 0, stream>>>(gsum, Wsm, summary, G, HID);
    k_bias_act <<<cdiv_i(G * HID, B), B, 0, stream>>>(summary, bsm, G * HID, 0);
}